// SparseGlobalBroadcast_27762668601803
// MI455X (gfx1250) — compile-verified
//
#include <hip/hip_runtime.h>
#include <stdint.h>

typedef float v4f __attribute__((ext_vector_type(4)));
typedef int   v4i __attribute__((ext_vector_type(4)));

#define CCH 128   // channels (reference: C = 128)
#define BB  8     // batch size (reference: B = 8)

// One wave32 processes FOUR consecutive sparse rows per loop iteration:
//   lane l covers channels [4l, 4l+3]  (32 lanes * float4 = 128 channels)
// The 4 batch indices are one aligned b128 load (wave-uniform).
// glob_feat (8x128 f32 = 4 KB) lives in LDS, preloaded via the CDNA5
// async global->LDS copy path (ASYNCcnt + s_wait_asynccnt).
__global__ __launch_bounds__(256) void sgb_add_kernel(
    const float* __restrict__ in_feat,
    const float* __restrict__ glob_feat,
    const int*   __restrict__ batch_idx,
    float*       __restrict__ out,
    int          n_rows)
{
    __shared__ float lds_glob[BB * CCH];   // 4096 bytes

    // ---- async preload of the 8x128 global-feature table into LDS ----
    // 256 threads x 16 B = 4096 B, one GLOBAL_LOAD_ASYNC_TO_LDS_B128 each.
    {
        uint32_t lds_addr = (uint32_t)(uintptr_t)(&lds_glob[0]) + threadIdx.x * 16u;
        uint32_t voff     = threadIdx.x * 16u;
        asm volatile("global_load_async_to_lds_b128 %0, %1, %2"
                     :
                     : "v"(lds_addr), "v"(voff), "s"(glob_feat)
                     : "memory");
        __builtin_amdgcn_s_wait_asynccnt(0);   // wait for own async copies
    }
    __syncthreads();                            // table visible to all waves

    const int lane  = (int)(threadIdx.x & 31u);
    const int gwave = (int)((blockIdx.x * blockDim.x + threadIdx.x) >> 5);
    const int nwave = (int)((gridDim.x * blockDim.x) >> 5);

    const v4f* __restrict__ in4  = (const v4f*)in_feat;
    v4f*       __restrict__ out4 = (v4f*)out;

    // ---- main loop: 4 rows (2 KB in, 2 KB out) per wave per iteration ----
    const int ngrp = n_rows >> 2;               // groups of 4 rows
    for (int grp = gwave; grp < ngrp; grp += nwave) {
        const int r0 = grp << 2;

        // 4 wave-uniform indices in one aligned 16 B load
        const v4i b4 = *(const v4i*)&batch_idx[r0];

        const size_t e0 = (size_t)r0 * 32u + (size_t)lane;
        // issue all 4 stream loads before any use (4 x b128 NT in flight)
        v4f x0 = __builtin_nontemporal_load(&in4[e0]);
        v4f x1 = __builtin_nontemporal_load(&in4[e0 + 32]);
        v4f x2 = __builtin_nontemporal_load(&in4[e0 + 64]);
        v4f x3 = __builtin_nontemporal_load(&in4[e0 + 96]);

        // conflict-free ds_load_b128 gathers from the 4 KB LDS table
        const v4f g0 = *(const v4f*)&lds_glob[b4.x * CCH + lane * 4];
        const v4f g1 = *(const v4f*)&lds_glob[b4.y * CCH + lane * 4];
        const v4f g2 = *(const v4f*)&lds_glob[b4.z * CCH + lane * 4];
        const v4f g3 = *(const v4f*)&lds_glob[b4.w * CCH + lane * 4];

        x0 += g0;  x1 += g1;  x2 += g2;  x3 += g3;

        __builtin_nontemporal_store(x0, &out4[e0]);
        __builtin_nontemporal_store(x1, &out4[e0 + 32]);
        __builtin_nontemporal_store(x2, &out4[e0 + 64]);
        __builtin_nontemporal_store(x3, &out4[e0 + 96]);
    }

    // ---- tail: up to 3 leftover rows, one row per wave ----
    for (int row = (ngrp << 2) + gwave; row < n_rows; row += nwave) {
        const int b = batch_idx[row];
        const v4f g = *(const v4f*)&lds_glob[b * CCH + lane * 4];
        const size_t e = (size_t)row * 32u + (size_t)lane;
        v4f x = __builtin_nontemporal_load(&in4[e]);
        x += g;
        __builtin_nontemporal_store(x, &out4[e]);
    }
}

extern "C" void kernel_launch(void* const* d_in, const int* in_sizes, int n_in,
                              void* d_out, int out_size, void* d_ws, size_t ws_size,
                              hipStream_t stream) {
    const float* in_feat = (const float*)d_in[0];   // [N, 128] f32
    const float* glob    = (const float*)d_in[1];   // [8, 128]  f32
    const int*   bidx    = (const int*)d_in[2];     // [N]       i32
    float*       out     = (float*)d_out;           // [N, 128]  f32

    const int n_rows = in_sizes[2];                 // N

    // 256 threads = 8 wave32 per block, 4 rows per wave per iteration.
    int blocks = (n_rows + 31) / 32;
    if (blocks > 4096) blocks = 4096;               // grid-stride beyond this
    if (blocks < 1)    blocks = 1;

    sgb_add_kernel<<<blocks, 256, 0, stream>>>(in_feat, glob, bidx, out, n_rows);
}